// RMSNormFuseQuantTl_54709293416650
// MI455X (gfx1250) — compile-verified
//
#include <hip/hip_runtime.h>
#include <stdint.h>

// ---------------- problem constants (match reference) ----------------
#define D        4096
#define RANK     256
#define QCOLS    (D - RANK)          // 3840
#define THREADS  256                 // 8 wave32 waves
#define VPT      (D / THREADS)       // 16 floats per thread
#define NWAVES   (THREADS / 32)
#define ROWS_PER_BLOCK 4
#define EPSF     1e-5f
#define QMAXF    127.0f

typedef unsigned int u32x4 __attribute__((ext_vector_type(4)));
typedef int          i32x4 __attribute__((ext_vector_type(4)));
typedef int          i32x8 __attribute__((ext_vector_type(8)));

// LDS (addrspace 3) typedef so we can recover the raw LDS byte offset for the D#.
typedef __attribute__((address_space(3))) float lds_f32;

// Issue a TDM load of one 4096-float row (16 KB) from global -> LDS.
// D# layout per CDNA5 ISA:
//  group0: [1:0]=count(1), [63:32]=lds_addr, [120:64]=global_addr, [127:126]=type(2)
//  group1: [17:16]=data_size(2 => 4B), tensor_dim0=4096, tensor_dim1=1,
//          tile_dim0=4096, tile_dim1=1, tensor_dim0_stride=4096
__device__ __forceinline__ void tdm_load_row(uint64_t gaddr, uint32_t lds_off) {
  u32x4 g0;
  g0.x = 1u;                                     // count = 1 (valid), is_restore=0, gather off
  g0.y = lds_off;                                // lds_addr (bytes)
  g0.z = (uint32_t)(gaddr & 0xffffffffu);        // global_addr[31:0]
  g0.w = (uint32_t)(gaddr >> 32) | (2u << 30);   // global_addr[56:32] | type=2

  i32x8 g1;
  g1[0] = (int)(2u << 16);                       // workgroup_mask=0, data_size=2 (4 bytes)
  g1[1] = (int)((uint32_t)D << 16);              // atomic_barrier_addr=0 | tensor_dim0[15:0]
  g1[2] = (int)(1u << 16);                       // tensor_dim0[31:16]=0 | tensor_dim1[15:0]=1
  g1[3] = (int)((uint32_t)D << 16);              // tensor_dim1[31:16]=0 | tile_dim0=4096
  g1[4] = 1;                                     // tile_dim1=1 | tile_dim2=0
  g1[5] = D;                                     // tensor_dim0_stride[31:0]
  g1[6] = 0;                                     // stride0[47:32] | stride1[15:0]
  g1[7] = 0;                                     // stride1[47:16]

  i32x4 z4 = {0, 0, 0, 0};
#if __clang_major__ >= 23
  i32x8 z8 = {0, 0, 0, 0, 0, 0, 0, 0};
  __builtin_amdgcn_tensor_load_to_lds(g0, g1, z4, z4, z8, 0);   // amdgpu-toolchain 6-arg form
#else
  __builtin_amdgcn_tensor_load_to_lds(g0, g1, z4, z4, 0);       // ROCm 7.2 5-arg form
#endif
}

__global__ __launch_bounds__(THREADS)
void rmsnorm_quant_tdm_kernel(const float* __restrict__ x,
                              float* __restrict__ out_r,
                              int8_t* __restrict__ out_q,
                              float* __restrict__ out_s,
                              int nrows) {
  __shared__ float buf[2][D];            // 32 KB double buffer (TDM destination)
  __shared__ float red_ss[NWAVES];
  __shared__ float red_am[NWAVES];

  const int t    = threadIdx.x;
  const int lane = t & 31;
  const int wave = t >> 5;
  const int row0 = blockIdx.x * ROWS_PER_BLOCK;

  // Prologue: DMA row0 into buffer 0 (wave 0 only; TDM ignores EXEC, one issue per wave).
  if (t == 0) {
    tdm_load_row((uint64_t)(uintptr_t)(x + (size_t)row0 * D),
                 (uint32_t)(uintptr_t)(lds_f32*)&buf[0][0]);
  }

  for (int i = 0; i < ROWS_PER_BLOCK; ++i) {
    const int row = row0 + i;
    if (t == 0) __builtin_amdgcn_s_wait_tensorcnt(0);   // row i resident in LDS
    __syncthreads();                                    // publish to all waves
    // Pipeline: kick off row i+1 into the other buffer while we compute row i.
    if (t == 0 && (i + 1) < ROWS_PER_BLOCK && (row + 1) < nrows) {
      tdm_load_row((uint64_t)(uintptr_t)(x + (size_t)(row + 1) * D),
                   (uint32_t)(uintptr_t)(lds_f32*)&buf[(i + 1) & 1][0]);
    }

    if (row < nrows) {
      const int bsel = i & 1;
      // ---- single pass: load 16 floats/thread from LDS, accumulate ss and amax ----
      float4 v[VPT / 4];
      float ss = 0.f, am = 0.f;
#pragma unroll
      for (int k = 0; k < VPT / 4; ++k) {
        v[k] = ((const float4*)&buf[bsel][t * VPT])[k];   // ds_load_b128
        ss += v[k].x * v[k].x + v[k].y * v[k].y + v[k].z * v[k].z + v[k].w * v[k].w;
      }
      if (t >= RANK / VPT) {  // threads 16..255 own cols >= 256 (quantized region)
#pragma unroll
        for (int k = 0; k < VPT / 4; ++k) {
          am = fmaxf(am, fmaxf(fmaxf(fabsf(v[k].x), fabsf(v[k].y)),
                               fmaxf(fabsf(v[k].z), fabsf(v[k].w))));
        }
      }
      // ---- wave32 butterfly reduction, then 8-wave LDS reduction ----
#pragma unroll
      for (int off = 16; off > 0; off >>= 1) {
        ss += __shfl_xor(ss, off, 32);
        am = fmaxf(am, __shfl_xor(am, off, 32));
      }
      if (lane == 0) { red_ss[wave] = ss; red_am[wave] = am; }
      __syncthreads();
      float sst = 0.f, amt = 0.f;
#pragma unroll
      for (int w = 0; w < NWAVES; ++w) {
        sst += red_ss[w];
        amt = fmaxf(amt, red_am[w]);
      }
      // amax(|y_q|) = r * amax(|x_q|) since r = rsqrt(ms+eps) > 0
      const float r     = rsqrtf(sst * (1.0f / (float)D) + EPSF);
      const float scale = amt * r * (1.0f / QMAXF);
      const float inv   = (scale > 0.f) ? (1.0f / scale) : 1.0f;
      const float s     = r * inv;    // maps raw x directly to quant domain

      if (t == 0) out_s[row] = scale;

      if (t < RANK / VPT) {
        // x_r output: y = x * r, 16 floats -> 4x b128 stores
        float4* o = (float4*)(out_r + (size_t)row * RANK + t * VPT);
#pragma unroll
        for (int k = 0; k < VPT / 4; ++k) {
          float4 y;
          y.x = v[k].x * r; y.y = v[k].y * r; y.z = v[k].z * r; y.w = v[k].w * r;
          o[k] = y;
        }
      } else {
        // int8 quant: round-to-nearest-even (rintf) matches jnp.round; clamp +-127
        uint32_t pk[VPT / 4];
#pragma unroll
        for (int k = 0; k < VPT / 4; ++k) {
          float q0 = fminf(fmaxf(rintf(v[k].x * s), -QMAXF), QMAXF);
          float q1 = fminf(fmaxf(rintf(v[k].y * s), -QMAXF), QMAXF);
          float q2 = fminf(fmaxf(rintf(v[k].z * s), -QMAXF), QMAXF);
          float q3 = fminf(fmaxf(rintf(v[k].w * s), -QMAXF), QMAXF);
          pk[k] = ((uint32_t)(uint8_t)(int8_t)(int)q0) |
                  ((uint32_t)(uint8_t)(int8_t)(int)q1 << 8) |
                  ((uint32_t)(uint8_t)(int8_t)(int)q2 << 16) |
                  ((uint32_t)(uint8_t)(int8_t)(int)q3 << 24);
        }
        // 16 bytes -> one b128 store; row*3840 and base are 16B aligned
        int4* o = (int4*)(out_q + (size_t)row * QCOLS + (size_t)(t - RANK / VPT) * VPT);
        *o = make_int4((int)pk[0], (int)pk[1], (int)pk[2], (int)pk[3]);
      }
    }
    __syncthreads();   // all reads of buf[i&1] done before it is re-targeted next iter
  }
}

extern "C" void kernel_launch(void* const* d_in, const int* in_sizes, int n_in,
                              void* d_out, int out_size, void* d_ws, size_t ws_size,
                              hipStream_t stream) {
  const float* x = (const float*)d_in[0];
  const int nrows = in_sizes[0] / D;   // 4*4096 = 16384 rows

  // Outputs concatenated flat in return order: x_r (f32), x_quant (i8), x_scale (f32)
  char* base = (char*)d_out;
  float*  out_r = (float*)base;
  int8_t* out_q = (int8_t*)(base + (size_t)nrows * RANK * sizeof(float));
  float*  out_s = (float*)(base + (size_t)nrows * RANK * sizeof(float) +
                                  (size_t)nrows * QCOLS * sizeof(int8_t));

  const int blocks = (nrows + ROWS_PER_BLOCK - 1) / ROWS_PER_BLOCK;
  hipLaunchKernelGGL(rmsnorm_quant_tdm_kernel, dim3(blocks), dim3(THREADS), 0, stream,
                     x, out_r, out_q, out_s, nrows);
}